// hysteresis_89223650607170
// MI455X (gfx1250) — compile-verified
//
#include <hip/hip_runtime.h>
#include <cstdint>

namespace {

constexpr int kB       = 8;
constexpr int kH       = 1024;
constexpr int kW       = 1024;
constexpr int kN       = kH * kW;      // 1 << 20
constexpr int kMaxPlus = 2 * kN;       // background label / pad value
constexpr int kTile    = 32;
constexpr int kHalo    = kTile + 2;    // 34 (init kernel tile)
constexpr int kThreads = 256;          // 8 wave32 waves
constexpr int kIters   = 15;           // MAX_ITER; body is a fixed point once converged

// Widened tile for the iterated min-pool kernel: 34 rows x 40 cols, starting at
// column bx-4 so every 4-int (16B) chunk is aligned and either fully in-image
// or fully out-of-image (W and tile origins are multiples of 4).
constexpr int kTH     = 34;
constexpr int kTW     = 40;
constexpr int kChunks = kTH * (kTW / 4);  // 340 B128 transfers per block

// ---- CDNA5 async global -> LDS copies (gfx1250), with safe fallbacks -------
typedef int v4i __attribute__((ext_vector_type(4)));
typedef __attribute__((address_space(1))) int gas_int;
typedef __attribute__((address_space(3))) int las_int;
typedef __attribute__((address_space(1))) v4i gas_v4i;
typedef __attribute__((address_space(3))) v4i las_v4i;

__device__ __forceinline__ void async_ld_b32(const int* g, int* l) {
#if defined(__gfx1250__) && __has_builtin(__builtin_amdgcn_global_load_async_to_lds_b32)
  __builtin_amdgcn_global_load_async_to_lds_b32(
      (gas_int*)(uintptr_t)g, (las_int*)(uint32_t)(uintptr_t)l, 0, 0);
#else
  *l = *g;
#endif
}

__device__ __forceinline__ void async_ld_b128(const int* g, int* l) {
#if defined(__gfx1250__) && __has_builtin(__builtin_amdgcn_global_load_async_to_lds_b128)
  __builtin_amdgcn_global_load_async_to_lds_b128(
      (gas_v4i*)(uintptr_t)g, (las_v4i*)(uint32_t)(uintptr_t)l, 0, 0);
#else
  *(int4*)l = *(const int4*)g;
#endif
}

__device__ __forceinline__ void wait_async0() {
#if defined(__gfx1250__) && __has_builtin(__builtin_amdgcn_s_wait_asynccnt)
  __builtin_amdgcn_s_wait_asynccnt(0);
#elif defined(__gfx1250__)
  asm volatile("s_wait_asynccnt 0" ::: "memory");
#endif
}

__device__ __forceinline__ int adj_idx(int v) {
  int idx = v - (v >= kN ? kN : 0);
  idx = idx < 0 ? 0 : idx;
  return idx > (kN - 1) ? (kN - 1) : idx;
}

// ---- init: vertex labels from x, then masked 3x3 min -> lab0 ---------------
__global__ __launch_bounds__(kThreads) void k_init(const float* __restrict__ x,
                                                   int* __restrict__ lab) {
  __shared__ int tile[kHalo * kHalo];
  const int bx = blockIdx.x * kTile;
  const int by = blockIdx.y * kTile;
  const int b  = blockIdx.z;
  const int tid = threadIdx.x;
  const float* xb = x + (size_t)b * kN;

  for (int k = tid; k < kHalo * kHalo; k += kThreads) {
    int ty = k / kHalo, tx = k - ty * kHalo;
    int gy = by + ty - 1, gx = bx + tx - 1;
    int v = kMaxPlus;  // pad value for min-pool
    if ((unsigned)gy < (unsigned)kH && (unsigned)gx < (unsigned)kW) {
      int pix  = gy * kW + gx;
      float xv = xb[pix];
      // vertex: background -> 2n, weak edge -> pix, strong edge -> n + pix
      v = (xv == 0.0f) ? kMaxPlus : ((xv > 1.0f) ? (kN + pix) : pix);
    }
    tile[k] = v;
  }
  __syncthreads();

  int* labb = lab + (size_t)b * kN;
#pragma unroll
  for (int j = 0; j < (kTile * kTile) / kThreads; ++j) {
    int oi = tid + j * kThreads;
    int oy = oi >> 5, ox = oi & 31;
    int c = tile[(oy + 1) * kHalo + (ox + 1)];
    int out = kMaxPlus;
    if (c < kMaxPlus) {  // edge pixel
      int m = c;
#pragma unroll
      for (int dy = 0; dy < 3; ++dy)
#pragma unroll
        for (int dx = 0; dx < 3; ++dx) m = min(m, tile[(oy + dy) * kHalo + (ox + dx)]);
      out = m;
    }
    labb[(by + oy) * kW + (bx + ox)] = out;
  }
}

// ---- step 1: x2 = fmn(lab); x1 = edge ? lab[adj(x2)] : lab; write x1 twice -
__global__ __launch_bounds__(kThreads) void k_minpool_gather(const int* __restrict__ lab,
                                                             int* __restrict__ x1a,
                                                             int* __restrict__ x1b) {
  __shared__ alignas(16) int tile[kTH * kTW];
  const int bx = blockIdx.x * kTile;
  const int by = blockIdx.y * kTile;
  const int b  = blockIdx.z;
  const int tid = threadIdx.x;
  const size_t bbase = (size_t)b * kN;
  const int* labb = lab + bbase;

  // Fill 34x40 tile with B128 async DMA; chunk c covers global cols
  // [bx-4+4*cx, bx+4*cx) of row by-1+ty. Chunks are 16B-aligned in both
  // global and LDS, and are always fully in- or fully out-of-image.
  for (int c = tid; c < kChunks; c += kThreads) {
    int ty  = c / (kTW / 4);
    int cx  = c - ty * (kTW / 4);
    int gy  = by + ty - 1;
    int gc0 = bx + 4 * cx - 4;
    int* dst = &tile[ty * kTW + 4 * cx];
    if ((unsigned)gy < (unsigned)kH && (unsigned)gc0 < (unsigned)(kW - 3)) {
      async_ld_b128(&labb[gy * kW + gc0], dst);  // async DMA into LDS
    } else {
      int4 pad = {kMaxPlus, kMaxPlus, kMaxPlus, kMaxPlus};
      *(int4*)dst = pad;
    }
  }
  wait_async0();
  __syncthreads();

#pragma unroll
  for (int j = 0; j < (kTile * kTile) / kThreads; ++j) {
    int oi = tid + j * kThreads;
    int oy = oi >> 5, ox = oi & 31;
    // output pixel (by+oy, bx+ox) -> tile row oy+1, tile col ox+4
    int li = tile[(oy + 1) * kTW + (ox + 4)];  // lab at this pixel
    int out = li;                              // background keeps 2n
    if (li < kMaxPlus) {
      int m = li;
#pragma unroll
      for (int dy = 0; dy < 3; ++dy)
#pragma unroll
        for (int dx = 0; dx < 3; ++dx) m = min(m, tile[(oy + dy) * kTW + (ox + 3 + dx)]);
      out = labb[adj_idx(m)];                  // pointer-jump gather
    }
    int gi = (by + oy) * kW + (bx + ox);
    x1a[bbase + gi] = out;
    x1b[bbase + gi] = out;
  }
}

// ---- step 2: tree hook: scatter-min vals=x1 at adj(lab_old) into dst(=x1) --
__global__ __launch_bounds__(kThreads) void k_hook(const int* __restrict__ lab_old,
                                                   const int* __restrict__ vals,
                                                   int* __restrict__ dst) {
  int i = blockIdx.x * kThreads + threadIdx.x;
  if (i >= kB * kN) return;
  int li = lab_old[i];
  if (li < kMaxPlus) {  // edge pixels only; background scatter is a no-op
    int base = i & ~(kN - 1);
    atomicMin(&dst[base + adj_idx(li)], vals[i]);
  }
}

// ---- step 3: regular pointer jump: lab = edge ? src[adj(src)] : src --------
__global__ __launch_bounds__(kThreads) void k_jump(const int* __restrict__ src,
                                                   int* __restrict__ dst) {
  int i = blockIdx.x * kThreads + threadIdx.x;
  if (i >= kB * kN) return;
  int l = src[i];
  if (l < kMaxPlus) {
    int base = i & ~(kN - 1);
    l = src[base + adj_idx(l)];
  }
  dst[i] = l;
}

// ---- finalize: int labels -> float output ----------------------------------
__global__ __launch_bounds__(kThreads) void k_out(const int* __restrict__ lab,
                                                  float* __restrict__ out) {
  int i = blockIdx.x * kThreads + threadIdx.x;
  if (i >= kB * kN) return;
  out[i] = (float)lab[i];
}

}  // namespace

extern "C" void kernel_launch(void* const* d_in, const int* in_sizes, int n_in,
                              void* d_out, int out_size, void* d_ws, size_t ws_size,
                              hipStream_t stream) {
  const float* x = (const float*)d_in[0];

  // Scratch layout: lab + lab2 in d_ws (64 MB); x1 values buffer reuses d_out
  // (32 MB of int scratch, fully overwritten by k_out at the end).
  int* lab  = (int*)d_ws;
  int* lab2 = lab + (size_t)kB * kN;
  int* tmp  = (int*)d_out;

  dim3 tgrid(kW / kTile, kH / kTile, kB);  // 32 x 32 x 8 tiles
  dim3 tblk(kThreads);
  const int total   = kB * kN;
  const int eblocks = (total + kThreads - 1) / kThreads;

  k_init<<<tgrid, tblk, 0, stream>>>(x, lab);

  // Fixed 15 iterations: the loop body is a fixed point once converged, so
  // this is exactly equivalent to the reference's early-exit while_loop and
  // keeps the launch sequence deterministic for graph capture.
  for (int it = 0; it < kIters; ++it) {
    k_minpool_gather<<<tgrid, tblk, 0, stream>>>(lab, tmp, lab2);
    k_hook<<<eblocks, tblk, 0, stream>>>(lab, tmp, lab2);
    k_jump<<<eblocks, tblk, 0, stream>>>(lab2, lab);
  }

  k_out<<<eblocks, tblk, 0, stream>>>(lab, (float*)d_out);
}